// Qwen2TransformerBlockWithCache_61555471286298
// MI455X (gfx1250) — compile-verified
//
#include <hip/hip_runtime.h>
#include <hip/hip_bf16.h>
#include <math.h>

typedef float v2f __attribute__((ext_vector_type(2)));
typedef float v8f __attribute__((ext_vector_type(8)));

// ---- problem constants (from reference setup_inputs) ----
constexpr int kB   = 8;
constexpr int kL   = 16;
constexpr int kE   = 2048;
constexpr int kH   = 16;
constexpr int kHKV = 2;
constexpr int kD   = 128;
constexpr int kG   = kH / kHKV;   // 8
constexpr int kF   = 5632;
constexpr int kBL  = kB * kL;     // 128 rows in every activation GEMM

// ============================================================
// RMSNorm: one block per row, 256 threads, LDS reduction
// ============================================================
__global__ void rmsnorm_kernel(const float* __restrict__ x,
                               const float* __restrict__ w,
                               float* __restrict__ out, int ncols) {
  int row = blockIdx.x;
  int tid = threadIdx.x;
  const float* xr = x + (long)row * ncols;
  float* orow = out + (long)row * ncols;
  __shared__ float red[256];
  float s = 0.f;
  for (int i = tid; i < ncols; i += 256) { float v = xr[i]; s += v * v; }
  red[tid] = s; __syncthreads();
  for (int k = 128; k > 0; k >>= 1) {
    if (tid < k) red[tid] += red[tid + k];
    __syncthreads();
  }
  float scale = rsqrtf(red[0] / (float)ncols + 1e-6f);
  for (int i = tid; i < ncols; i += 256) orow[i] = xr[i] * scale * w[i];
}

// ============================================================
// Generic WMMA GEMM: Out[m][n] = sum_k A[m][k]*W[n][k] (+bias[n]) (+resid[m][n])
// A: (kBL=128 x K) row-major, W: (N x K) row-major.
// One wave computes a 64(M) x 16(N) strip = 4 M-tiles, reusing the weight
// B-fragment 4x per k-step. Per 16-float body: prefetch, then ALL 20
// fragment loads into distinct temps (clause-batched, staged waits), then
// 16 WMMAs. job = tn*2 + mg so wave pairs share the weight rows.
// ============================================================
__global__ void gemm_wmma_kernel(const float* __restrict__ A,
                                 const float* __restrict__ W,
                                 const float* __restrict__ bias,
                                 const float* __restrict__ resid,
                                 float* __restrict__ Out,
                                 int N, int K, int totalJobs) {
  int lane = threadIdx.x & 31;
  int wave = threadIdx.x >> 5;
  int job = blockIdx.x * (blockDim.x >> 5) + wave;
  if (job >= totalJobs) return;
  int mg = job & 1;   // 64-row group of A (kBL = 128)
  int tn = job >> 1;  // N tile

  int ln = lane & 15;
  int kh = lane >> 4;
  int koff = kh * 2;

  const float* brow = W + (long)(tn * 16 + ln) * K + koff;
  const float* a0   = A + (long)(mg * 64 + ln) * K + koff;
  const long aStr = 16L * K;

  v8f acc[4] = {{}, {}, {}, {}};
  for (int kk = 0; kk < K; kk += 16) {
    __builtin_prefetch(brow + kk + 1024, 0, 3);  // next weight bytes
    v2f bf[4];
    v2f af[4][4];
#pragma unroll
    for (int j = 0; j < 4; ++j) {
      int k = kk + 4 * j;
      bf[j] = *(const v2f*)(brow + k);
#pragma unroll
      for (int t = 0; t < 4; ++t)
        af[j][t] = *(const v2f*)(a0 + t * aStr + k);
    }
#pragma unroll
    for (int j = 0; j < 4; ++j) {
#pragma unroll
      for (int t = 0; t < 4; ++t) {
        acc[t] = __builtin_amdgcn_wmma_f32_16x16x4_f32(false, af[j][t], false,
                                                       bf[j], (short)0, acc[t],
                                                       false, false);
      }
    }
  }

  int n = tn * 16 + ln;
  float bval = bias ? bias[n] : 0.f;
#pragma unroll
  for (int t = 0; t < 4; ++t) {
#pragma unroll
    for (int i = 0; i < 8; ++i) {
      int m = mg * 64 + t * 16 + kh * 8 + i;
      float r = acc[t][i] + bval;
      if (resid) r += resid[(long)m * N + n];
      Out[(long)m * N + n] = r;
    }
  }
}

// ============================================================
// RoPE (HF half-split) in place on (kB, kL, heads, kD)
// ============================================================
__global__ void rope_kernel(float* __restrict__ x, const int* __restrict__ offp,
                            int heads) {
  long idx = (long)blockIdx.x * blockDim.x + threadIdx.x;
  long total = (long)kB * kL * heads * (kD / 2);
  if (idx >= total) return;
  int d = idx & 63; long t = idx >> 6;
  int h = t % heads; t /= heads;
  int l = t % kL; int b = (int)(t / kL);

  float pos = (float)(*offp + l);
  float invf = __expf(-(float)d * (logf(10000.0f) / 64.0f));
  float ang = pos * invf;
  float c = cosf(ang), s = sinf(ang);

  float* base = x + ((long)(b * kL + l) * heads + h) * kD;
  float x0 = base[d], x1 = base[d + 64];
  base[d]      = x0 * c - x1 * s;
  base[d + 64] = x1 * c + x0 * s;
}

// ============================================================
// Attention scores: for each (b,kvh,g): scores[l][s] = (q . K_s)*scale, masked.
// One wave computes 16(L) x 64(S) = 4 S-tiles, reusing the query A-fragment 4x.
// S tiles handled with per-tile validity for the 257-tile (S=4112) tail.
// ============================================================
__global__ void attn_scores_kernel(const float* __restrict__ q,
                                   const float* __restrict__ cache_k,
                                   const float* __restrict__ knew,
                                   const int* __restrict__ offp,
                                   float* __restrict__ scores,
                                   int S, int S0) {
  int lane = threadIdx.x & 31;
  int wave = threadIdx.x >> 5;
  int tilesS = S / 16;
  int s4cnt = (tilesS + 3) / 4;
  long job = (long)blockIdx.x * (blockDim.x >> 5) + wave;
  long total = (long)kB * kHKV * kG * s4cnt;
  if (job >= total) return;

  int s4 = (int)(job % s4cnt); long t2 = job / s4cnt;
  int g = (int)(t2 % kG); t2 /= kG;
  int kvh = (int)(t2 % kHKV); int b = (int)(t2 / kHKV);

  int ln = lane & 15;
  int kh = lane >> 4;
  int koff = kh * 2;

  const float* arow = q + ((long)(b * kL + ln) * kH + kvh * kG + g) * kD + koff;

  const float* brow[4];
  bool valid[4];
#pragma unroll
  for (int t = 0; t < 4; ++t) {
    int ts = s4 * 4 + t;
    valid[t] = ts < tilesS;
    int s = ts * 16 + ln;
    if (!valid[t]) {
      brow[t] = cache_k;  // safe dummy
    } else if (s < S0) {
      brow[t] = cache_k + ((long)(b * kHKV + kvh) * S0 + s) * kD;
    } else {
      brow[t] = knew + ((long)(b * kL + (s - S0)) * kHKV + kvh) * kD;
    }
    brow[t] += koff;
  }

  v8f acc[4] = {{}, {}, {}, {}};
#pragma unroll
  for (int kk = 0; kk < kD; kk += 4) {
    v2f av = *(const v2f*)(arow + kk);
#pragma unroll
    for (int t = 0; t < 4; ++t) {
      v2f bv = *(const v2f*)(brow[t] + kk);
      acc[t] = __builtin_amdgcn_wmma_f32_16x16x4_f32(false, av, false, bv,
                                                     (short)0, acc[t], false, false);
    }
  }

  int off = *offp;
  const float sc = 0.08838834764831845f;  // 1/sqrt(128)
  float* obase = scores + ((long)((b * kHKV + kvh) * (kG * kL) + g * kL)) * S;
#pragma unroll
  for (int t = 0; t < 4; ++t) {
    if (!valid[t]) continue;
    int n = (s4 * 4 + t) * 16 + ln;  // key position for this lane's column
#pragma unroll
    for (int i = 0; i < 8; ++i) {
      int l = kh * 8 + i;  // query position within block
      float val = acc[t][i] * sc;
      if (n > off + l) val = -1e30f;
      obase[(long)l * S + n] = val;
    }
  }
}

// ============================================================
// Row softmax over S (block per row)
// ============================================================
__global__ void softmax_kernel(float* __restrict__ scores, int S) {
  int row = blockIdx.x;
  int tid = threadIdx.x;
  float* p = scores + (long)row * S;
  __shared__ float red[256];

  float m = -1e30f;
  for (int i = tid; i < S; i += 256) m = fmaxf(m, p[i]);
  red[tid] = m; __syncthreads();
  for (int k = 128; k > 0; k >>= 1) {
    if (tid < k) red[tid] = fmaxf(red[tid], red[tid + k]);
    __syncthreads();
  }
  float mx = red[0]; __syncthreads();

  float sum = 0.f;
  for (int i = tid; i < S; i += 256) {
    float e = __expf(p[i] - mx);
    p[i] = e;
    sum += e;
  }
  red[tid] = sum; __syncthreads();
  for (int k = 128; k > 0; k >>= 1) {
    if (tid < k) red[tid] += red[tid + k];
    __syncthreads();
  }
  float inv = 1.0f / red[0];
  for (int i = tid; i < S; i += 256) p[i] *= inv;
}

// ============================================================
// o = P @ V : per (b,kvh), M=128 rows (g*16+l), N=kD, K=S.
// One wave computes 4 g-tiles for one 16-wide d-tile, reusing the
// column-gathered V fragment 4x. Per 16-step body: prefetch, 24 batched
// loads into distinct temps, then 16 WMMAs. Output written into (b,l,h,d).
// ============================================================
__global__ void attn_v_kernel(const float* __restrict__ scores,
                              const float* __restrict__ cache_v,
                              const float* __restrict__ vnew,
                              float* __restrict__ o,
                              int S, int S0) {
  int lane = threadIdx.x & 31;
  int wave = threadIdx.x >> 5;
  constexpr int tilesN = kD / 16;  // 8
  long job = (long)blockIdx.x * (blockDim.x >> 5) + wave;
  long total = (long)kB * kHKV * 2 * tilesN;  // 2 groups of 4 g each
  if (job >= total) return;

  int tn = (int)(job % tilesN); long t2 = job / tilesN;
  int gp = (int)(t2 & 1); t2 >>= 1;
  int kvh = (int)(t2 % kHKV); int b = (int)(t2 / kHKV);

  int ln = lane & 15;
  int kh = lane >> 4;
  int koff = kh * 2;
  int d0 = tn * 16;

  const float* arow[4];
#pragma unroll
  for (int t = 0; t < 4; ++t) {
    int g = gp * 4 + t;
    arow[t] = scores +
        ((long)((b * kHKV + kvh) * (kG * kL) + g * kL + ln)) * S + koff;
  }
  const float* vc = cache_v + (long)(b * kHKV + kvh) * S0 * kD + d0 + ln;

  v8f acc[4] = {{}, {}, {}, {}};
  // cached part of the K(=S) dimension
  for (int kk = 0; kk < S0; kk += 16) {
    __builtin_prefetch(vc + (long)(kk + koff + 64) * kD, 0, 3);
    __builtin_prefetch(arow[0] + kk + 1024, 0, 3);
    v2f bf[4];
    v2f af[4][4];
#pragma unroll
    for (int j = 0; j < 4; ++j) {
      long sr = kk + 4 * j + koff;
      bf[j].x = vc[sr * kD];
      bf[j].y = vc[(sr + 1) * kD];
#pragma unroll
      for (int t = 0; t < 4; ++t)
        af[j][t] = *(const v2f*)(arow[t] + kk + 4 * j);
    }
#pragma unroll
    for (int j = 0; j < 4; ++j) {
#pragma unroll
      for (int t = 0; t < 4; ++t) {
        acc[t] = __builtin_amdgcn_wmma_f32_16x16x4_f32(false, af[j][t], false,
                                                       bf[j], (short)0, acc[t],
                                                       false, false);
      }
    }
  }
  // new-token part
  for (int kk = S0; kk < S; kk += 4) {
    int l2 = kk + koff - S0;
    v2f bv;
    bv.x = vnew[((long)(b * kL + l2) * kHKV + kvh) * kD + d0 + ln];
    bv.y = vnew[((long)(b * kL + l2 + 1) * kHKV + kvh) * kD + d0 + ln];
#pragma unroll
    for (int t = 0; t < 4; ++t) {
      v2f av = *(const v2f*)(arow[t] + kk);
      acc[t] = __builtin_amdgcn_wmma_f32_16x16x4_f32(false, av, false, bv,
                                                     (short)0, acc[t], false, false);
    }
  }

#pragma unroll
  for (int t = 0; t < 4; ++t) {
    int g = gp * 4 + t;
#pragma unroll
    for (int i = 0; i < 8; ++i) {
      int l = kh * 8 + i;
      o[((long)(b * kL + l) * kH + kvh * kG + g) * kD + d0 + ln] = acc[t][i];
    }
  }
}

// ============================================================
// act = silu(gate) * up, in place on gate
// ============================================================
__global__ void silu_mul_kernel(float* __restrict__ gate,
                                const float* __restrict__ up, long n) {
  long i = (long)blockIdx.x * blockDim.x + threadIdx.x;
  if (i < n) {
    float g = gate[i];
    gate[i] = (g / (1.f + __expf(-g))) * up[i];
  }
}

// ============================================================
extern "C" void kernel_launch(void* const* d_in, const int* in_sizes, int n_in,
                              void* d_out, int out_size, void* d_ws, size_t ws_size,
                              hipStream_t stream) {
  const float* x        = (const float*)d_in[0];
  const float* cache_k  = (const float*)d_in[1];
  const float* cache_v  = (const float*)d_in[2];
  const float* wq       = (const float*)d_in[3];
  const float* wk       = (const float*)d_in[4];
  const float* wv       = (const float*)d_in[5];
  const float* wo       = (const float*)d_in[6];
  const float* bq       = (const float*)d_in[7];
  const float* bk       = (const float*)d_in[8];
  const float* bv       = (const float*)d_in[9];
  const float* w_gate   = (const float*)d_in[10];
  const float* w_up     = (const float*)d_in[11];
  const float* w_down   = (const float*)d_in[12];
  const float* w_in_ln  = (const float*)d_in[13];
  const float* w_post_ln= (const float*)d_in[14];
  const int*   offp     = (const int*)d_in[15];

  const int S0 = in_sizes[1] / (kB * kHKV * kD);  // 4096
  const int S  = S0 + kL;                         // 4112 (multiple of 16)

  float* ws = (float*)d_ws;
  size_t off = 0;
  float* xn    = ws + off; off += (size_t)kBL * kE;
  float* qb    = ws + off; off += (size_t)kBL * kE;
  float* kb    = ws + off; off += (size_t)kBL * kHKV * kD;
  float* vb    = ws + off; off += (size_t)kBL * kHKV * kD;
  float* hb    = ws + off; off += (size_t)kBL * kE;
  float* hn    = ws + off; off += (size_t)kBL * kE;
  float* ob    = ws + off; off += (size_t)kBL * kE;
  float* gateb = ws + off; off += (size_t)kBL * kF;
  float* upb   = ws + off; off += (size_t)kBL * kF;
  float* sc    = ws + off; off += (size_t)kB * kHKV * kG * kL * S;

  auto launch_gemm = [&](const float* A, const float* W, const float* bias,
                         const float* resid, float* Out, int N, int K) {
    int total = 2 * (N / 16);  // 2 x 64-row M groups
    int blocks = (total + 7) / 8;
    gemm_wmma_kernel<<<blocks, 256, 0, stream>>>(A, W, bias, resid, Out, N, K,
                                                 total);
  };

  // 1. input rmsnorm
  rmsnorm_kernel<<<kBL, 256, 0, stream>>>(x, w_in_ln, xn, kE);

  // 2. q/k/v projections
  launch_gemm(xn, wq, bq, nullptr, qb, kE, kE);
  launch_gemm(xn, wk, bk, nullptr, kb, kHKV * kD, kE);
  launch_gemm(xn, wv, bv, nullptr, vb, kHKV * kD, kE);

  // 3. RoPE on q and new k
  {
    long tq = (long)kB * kL * kH * (kD / 2);
    rope_kernel<<<(int)((tq + 255) / 256), 256, 0, stream>>>(qb, offp, kH);
    long tk = (long)kB * kL * kHKV * (kD / 2);
    rope_kernel<<<(int)((tk + 255) / 256), 256, 0, stream>>>(kb, offp, kHKV);
  }

  // 4. scores = q @ K^T * scale, masked
  {
    int s4cnt = (S / 16 + 3) / 4;
    long total = (long)kB * kHKV * kG * s4cnt;
    int blocks = (int)((total + 7) / 8);
    attn_scores_kernel<<<blocks, 256, 0, stream>>>(qb, cache_k, kb, offp, sc, S, S0);
  }

  // 5. softmax over S
  softmax_kernel<<<kB * kHKV * kG * kL, 256, 0, stream>>>(sc, S);

  // 6. o = P @ V (written straight into (b,l,h,d))
  {
    long total = (long)kB * kHKV * 2 * (kD / 16);
    int blocks = (int)((total + 7) / 8);
    attn_v_kernel<<<blocks, 256, 0, stream>>>(sc, cache_v, vb, ob, S, S0);
  }

  // 7. h = x + o @ wo^T
  launch_gemm(ob, wo, nullptr, x, hb, kE, kE);

  // 8. post rmsnorm
  rmsnorm_kernel<<<kBL, 256, 0, stream>>>(hb, w_post_ln, hn, kE);

  // 9. MLP
  launch_gemm(hn, w_gate, nullptr, nullptr, gateb, kF, kE);
  launch_gemm(hn, w_up,   nullptr, nullptr, upb,   kF, kE);
  {
    long n = (long)kBL * kF;
    silu_mul_kernel<<<(int)((n + 255) / 256), 256, 0, stream>>>(gateb, upb, n);
  }
  // 10. out = h + act @ w_down^T  (written directly to d_out)
  launch_gemm(gateb, w_down, nullptr, hb, (float*)d_out, kE, kF);

  (void)n_in; (void)out_size; (void)ws_size;
}